// FeatureAdaption_50818053046399
// MI455X (gfx1250) — compile-verified
//
#include <hip/hip_runtime.h>
#include <hip/hip_bf16.h>

typedef __attribute__((ext_vector_type(2))) float v2f;
typedef __attribute__((ext_vector_type(4))) float v4f;
typedef __attribute__((ext_vector_type(8))) float v8f;
typedef __attribute__((ext_vector_type(4))) int   v4i;

#define BATCH 2
#define CIN   256
#define HH    64
#define WWID  64
#define HW    4096
#define DGN   4
#define CGRP  64          // CIN / DGN
#define KTAP  9
#define COUT  256
#define CKDIM 2304        // CIN * KTAP
#define CKG   576         // CGRP * KTAP (per-group contraction)
#define STEPG 144         // CKG / 4 (WMMA K=4 steps per group)
#define MT    32          // pixel tile (2 WMMA M-tiles)
#define APAD  580         // LDS row stride: even (b64 align), ==4 mod 64 banks

// ---------------------------------------------------------------------------
// Kernel 1: offsets (1x1 conv) + bilinear tap precompute.
// One thread per (b, g, k, pix). Tap record = {idx[4] (clamped), w[4] (masked)}.
// gid = ((b*4+g)*9 + k)*4096 + pix  -> taps[gid*8 .. gid*8+7]
// ---------------------------------------------------------------------------
__global__ __launch_bounds__(256)
void fa_tap_kernel(const float* __restrict__ pred,
                   const float* __restrict__ woff,
                   int* __restrict__ taps) {
  int gid = blockIdx.x * 256 + threadIdx.x;
  int pix = gid & (HW - 1);
  int e   = gid >> 12;          // (b*4+g)*9 + k
  int k   = e % KTAP;
  int bg  = e / KTAP;           // b*4 + g
  int b   = bg >> 2;
  int g   = bg & 3;
  int h   = pix >> 6;
  int w   = pix & 63;
  int ky  = k / 3;
  int kx  = k % 3;
  int row = (g * KTAP + k) * 2; // offset channel for y; +1 for x

  float oy = 0.f, ox = 0.f;
  const float* pp = pred + (size_t)b * 34 * HW + pix;
#pragma unroll
  for (int c = 0; c < 34; ++c) {
    float pv = pp[(size_t)c * HW];
    oy = fmaf(woff[row * 34 + c], pv, oy);
    ox = fmaf(woff[(row + 1) * 34 + c], pv, ox);
  }
  float py = oy + (float)(h + ky - 1);
  float px = ox + (float)(w + kx - 1);
  float y0 = floorf(py), x0 = floorf(px);
  float fy = py - y0,    fx = px - x0;
  int iy0 = (int)y0, ix0 = (int)x0;

  v4i idx; v4f wt;
#pragma unroll
  for (int c = 0; c < 4; ++c) {
    int dy = c >> 1, dx = c & 1;
    int yc = iy0 + dy, xc = ix0 + dx;
    float wy = dy ? fy : 1.f - fy;
    float wx = dx ? fx : 1.f - fx;
    bool valid = (yc >= 0) & (yc <= HH - 1) & (xc >= 0) & (xc <= WWID - 1);
    float ww = valid ? wy * wx : 0.f;
    int yl = min(max(yc, 0), HH - 1);
    int xl = min(max(xc, 0), WWID - 1);
    idx[c] = yl * WWID + xl;
    wt[c]  = ww;
  }
  int* tp = taps + (size_t)gid * 8;
  *(v4i*)(tp)     = idx;
  *(v4f*)(tp + 4) = wt;
}

// ---------------------------------------------------------------------------
// Kernel 2: permute weight [COUT, CKDIM] into WMMA-B fragment order.
// Wfrag index = ((s*16 + nt)*32 + lane)*2, s = ck/4, nt = cout tile,
// lane<16 -> K pair {0,1}, lane>=16 -> {2,3}; n = nt*16 + lane%16.
// ---------------------------------------------------------------------------
__global__ __launch_bounds__(256)
void fa_wfrag_kernel(const float* __restrict__ wsrc,
                     float* __restrict__ wfrag) {
  int gid  = blockIdx.x * 256 + threadIdx.x;   // 576*16*32 = 294912 total
  int lane = gid & 31;
  int nt   = (gid >> 5) & 15;
  int s    = gid >> 9;
  int n    = nt * 16 + (lane & 15);
  int k0   = s * 4 + ((lane < 16) ? 0 : 2);
  v2f o;
  o.x = wsrc[(size_t)n * CKDIM + k0];
  o.y = wsrc[(size_t)n * CKDIM + k0 + 1];
  *(v2f*)(wfrag + (size_t)gid * 2) = o;
}

// ---------------------------------------------------------------------------
// Kernel 3: fused im2col + WMMA GEMM + ReLU.
// grid.x = HW/MT pixel tiles, grid.y = conv*2 + b. 256 threads = 8 waves.
// Wave w owns cout tiles {2w, 2w+1} for BOTH 16-px M-tiles: 4 accumulators,
// per K-step: 2 ds_load_b64 (A) + 2 global_load_b64 (B) + 4 WMMA.
// ---------------------------------------------------------------------------
__global__ __launch_bounds__(256)
void fa_deform_wmma_kernel(const float* __restrict__ reg_feat,
                           const float* __restrict__ cls_feat,
                           const int*   __restrict__ taps,
                           const float* __restrict__ wfrag_reg,
                           const float* __restrict__ wfrag_cls,
                           float* __restrict__ out) {
  __shared__ float Atile[MT * APAD];          // 74,240 B
  __shared__ int   tapI[MT * KTAP * 8];       //  9,216 B

  const int tid  = threadIdx.x;
  const int lane = tid & 31;
  const int wave = tid >> 5;
  const int pix_base = blockIdx.x * MT;
  const int sel  = blockIdx.y;                // conv*2 + b
  const int conv = sel >> 1;
  const int b    = sel & 1;

  const float* x   = (conv ? cls_feat : reg_feat) + (size_t)b * CIN * HW;
  const float* wf  = conv ? wfrag_cls : wfrag_reg;
  float* outp = out + (size_t)conv * (BATCH * COUT * HW) + (size_t)b * COUT * HW;

  v8f acc00 = {}, acc01 = {}, acc10 = {}, acc11 = {};
  const int nt0   = wave * 2;                 // cout tiles 2w, 2w+1
  const int m     = lane & 15;                // A-matrix row within M-tile
  const int koffA = (lane < 16) ? 0 : 2;      // A fragment K offset

  for (int g = 0; g < DGN; ++g) {
    __syncthreads();                          // protect Atile/tapI reuse
    // Stage taps for (b, g): 288 records = 2304 dwords.
    const int* tsrc = taps + (size_t)(b * 4 + g) * KTAP * HW * 8;
    for (int t = tid; t < MT * KTAP * 8; t += 256) {
      int dw = t & 7;
      int e  = t >> 3;                        // k*32 + mm
      int k  = e >> 5;
      int mm = e & 31;
      tapI[t] = tsrc[((size_t)k * HW + (pix_base + mm)) * 8 + dw];
    }
    __syncthreads();
    // Build A tile [32 px][576]. Each unit = (pixel mm, tap k, 8 channels):
    // tap held in registers, 4 corner bases computed once, channel loop uses
    // constant +cg*HW immediates on the gathers.
#pragma unroll
    for (int j = 0; j < 9; ++j) {
      int unit  = j * 256 + tid;              // 2304 units
      int mm    = unit & 31;
      int rest  = unit >> 5;                  // 0..71
      int k     = rest >> 3;                  // 0..8
      int cg0   = (rest & 7) * 8;             // channel chunk base
      const int* tp = tapI + (k * 32 + mm) * 8;
      int   i0 = tp[0], i1 = tp[1], i2 = tp[2], i3 = tp[3];
      float w0 = __int_as_float(tp[4]), w1 = __int_as_float(tp[5]);
      float w2 = __int_as_float(tp[6]), w3 = __int_as_float(tp[7]);
      const float* xp = x + (size_t)(g * CGRP + cg0) * HW;
      const float* p0 = xp + i0;
      const float* p1 = xp + i1;
      const float* p2 = xp + i2;
      const float* p3 = xp + i3;
      float* arow = &Atile[mm * APAD + cg0 * KTAP + k];
#pragma unroll
      for (int cc = 0; cc < 8; ++cc) {
        float v = w0 * p0[cc * HW];
        v = fmaf(w1, p1[cc * HW], v);
        v = fmaf(w2, p2[cc * HW], v);
        v = fmaf(w3, p3[cc * HW], v);
        arow[cc * KTAP] = v;
      }
    }
    __syncthreads();
    // Contract this group's 576 K-slice: 144 steps x 4 WMMA (f32 16x16x4).
    const float* wbase = wf + (size_t)(g * STEPG) * 1024 + (nt0 * 32 + lane) * 2;
    const float* a0p = &Atile[m * APAD + koffA];
    const float* a1p = &Atile[(16 + m) * APAD + koffA];
#pragma unroll 4
    for (int s = 0; s < STEPG; ++s) {
      v2f af0 = *(const v2f*)(a0p + 4 * s);
      v2f af1 = *(const v2f*)(a1p + 4 * s);
      v2f b0  = *(const v2f*)(wbase + (size_t)s * 1024);
      v2f b1  = *(const v2f*)(wbase + (size_t)s * 1024 + 64);
      acc00 = __builtin_amdgcn_wmma_f32_16x16x4_f32(
          false, af0, false, b0, (short)0, acc00, false, false);
      acc01 = __builtin_amdgcn_wmma_f32_16x16x4_f32(
          false, af0, false, b1, (short)0, acc01, false, false);
      acc10 = __builtin_amdgcn_wmma_f32_16x16x4_f32(
          false, af1, false, b0, (short)0, acc10, false, false);
      acc11 = __builtin_amdgcn_wmma_f32_16x16x4_f32(
          false, af1, false, b1, (short)0, acc11, false, false);
    }
  }

  // ReLU + store. D layout: VGPR r -> pixel m = r + 8*(lane>=16), N = lane%16.
  const int mrow = (lane >> 4) * 8;
#pragma unroll
  for (int mt = 0; mt < 2; ++mt) {
#pragma unroll
    for (int t = 0; t < 2; ++t) {
      v8f a = mt ? (t ? acc11 : acc10) : (t ? acc01 : acc00);
      int n = (nt0 + t) * 16 + (lane & 15);
      float* op = outp + (size_t)n * HW + pix_base + mt * 16 + mrow;
      v4f lo, hi;
#pragma unroll
      for (int r = 0; r < 4; ++r) {
        lo[r] = fmaxf(a[r], 0.f);
        hi[r] = fmaxf(a[r + 4], 0.f);
      }
      *(v4f*)(op)     = lo;
      *(v4f*)(op + 4) = hi;
    }
  }
}

extern "C" void kernel_launch(void* const* d_in, const int* in_sizes, int n_in,
                              void* d_out, int out_size, void* d_ws, size_t ws_size,
                              hipStream_t stream) {
  const float* reg_feat = (const float*)d_in[0];
  const float* cls_feat = (const float*)d_in[1];
  const float* pred     = (const float*)d_in[2];
  const float* w_offset = (const float*)d_in[3];
  const float* w_reg    = (const float*)d_in[4];
  const float* w_cls    = (const float*)d_in[5];
  float* out = (float*)d_out;

  // Workspace layout: taps (9.44 MB) | wfrag_reg (2.25 MB) | wfrag_cls (2.25 MB)
  int*   taps      = (int*)d_ws;
  size_t tapDwords = (size_t)BATCH * DGN * KTAP * HW * 8;
  float* wfrag_reg = (float*)d_ws + tapDwords;
  float* wfrag_cls = wfrag_reg + (size_t)(CKDIM / 4) * 16 * 32 * 2;

  // 1) offsets + bilinear taps: 294,912 threads
  fa_tap_kernel<<<(BATCH * DGN * KTAP * HW) / 256, 256, 0, stream>>>(
      pred, w_offset, taps);
  // 2) weight fragment permutation (both convs)
  fa_wfrag_kernel<<<((CKDIM / 4) * 16 * 32) / 256, 256, 0, stream>>>(w_reg, wfrag_reg);
  fa_wfrag_kernel<<<((CKDIM / 4) * 16 * 32) / 256, 256, 0, stream>>>(w_cls, wfrag_cls);
  // 3) fused deformable conv + GEMM + ReLU
  dim3 grid(HW / MT, BATCH * 2);
  fa_deform_wmma_kernel<<<grid, 256, 0, stream>>>(
      reg_feat, cls_feat, taps, wfrag_reg, wfrag_cls, out);
}